// FinalTextConditionedOutput_53704271069781
// MI455X (gfx1250) — compile-verified
//
#include <hip/hip_runtime.h>
#include <hip/hip_bf16.h>
#include <math.h>

// Sizes from the reference
#define Bsz 256
#define Tsz 512
#define Dsz 768
#define Fsz 1024
#define Hsz 8
#define HIDsz 512
#define OUTsz 256
#define HEADD 128   // F/H
#define SCALE 0.08838834764831845f  // 1/sqrt(128)

typedef float v2f __attribute__((ext_vector_type(2)));
typedef float v8f __attribute__((ext_vector_type(8)));

// ---- fp32 WMMA 16x16x4 wrapper (CDNA5 V_WMMA_F32_16X16X4_F32) ----
// 8-arg group per probe doc: (neg_a, A, neg_b, B, c_mod, C, reuse_a, reuse_b)
__device__ __forceinline__ v8f wmma4(v2f a, v2f b, v8f c) {
  return __builtin_amdgcn_wmma_f32_16x16x4_f32(
      false, a, false, b, (short)0, c, false, false);
}

// A/B fragment gather for "row = lane&15, k + 2*(lane>>4)" pattern.
// (A 16x4: lanes 0-15 hold K=0,1; lanes 16-31 hold K=2,3 — ISA §7.12.2.
//  B 4x16 mirrored: N across lanes, K across VGPRs/lane-halves.)

// =====================================================================
// Kernel 1: per-batch prep.  q = x@Wq^T+bq;  qkp[b][dd][h16] = (Wk_h^T q_h)*scale
// (transposed layout [D x 16] for coalesced WMMA B-fragments); sbias = (q_h . bk_h)*scale
// =====================================================================
__global__ void prep_qk_kernel(const float* __restrict__ x,
                               const float* __restrict__ Wq, const float* __restrict__ bq,
                               const float* __restrict__ Wk, const float* __restrict__ bk,
                               float* __restrict__ qkp, float* __restrict__ sbias) {
  const int b = blockIdx.x, tid = threadIdx.x;
  __shared__ float xb[Fsz];
  __shared__ float qb[Fsz];
  for (int i = tid; i < Fsz; i += 256) xb[i] = x[(size_t)b * Fsz + i];
  __syncthreads();
  for (int f = tid; f < Fsz; f += 256) {
    const float* w = Wq + (size_t)f * Fsz;
    float acc = bq[f];
    for (int k = 0; k < Fsz; ++k) acc = fmaf(xb[k], w[k], acc);
    qb[f] = acc;
  }
  __syncthreads();
  // qk[h][dd] = sum_j Wk[h*128+j][dd] * q[h*128+j], scaled; store as qkp[dd][h]
  for (int h = 0; h < Hsz; ++h) {
    const float* qh = qb + h * HEADD;
    for (int dd = tid; dd < Dsz; dd += 256) {
      const float* wcol = Wk + (size_t)h * HEADD * Dsz + dd;
      float acc = 0.f;
      for (int j = 0; j < HEADD; ++j) acc = fmaf(wcol[(size_t)j * Dsz], qh[j], acc);
      qkp[(((size_t)b * Dsz) + dd) * 16 + h] = acc * SCALE;
    }
  }
  // zero the pad heads 8..15
  for (int idx = tid; idx < Dsz * 8; idx += 256) {
    int dd = idx >> 3, h = (idx & 7) + 8;
    qkp[(((size_t)b * Dsz) + dd) * 16 + h] = 0.f;
  }
  if (tid < Hsz) {
    float acc = 0.f;
    for (int j = 0; j < HEADD; ++j) acc = fmaf(qb[tid * HEADD + j], bk[tid * HEADD + j], acc);
    sbias[b * Hsz + tid] = acc * SCALE;
  }
}

// =====================================================================
// Kernel 2: per-batch fused scores + softmax + ctx (one block per batch, 8 waves)
//   pass1: S^T[Tsz x 16] = text[b] (T x D) @ qkp (D x 16)   — WMMA fp32
//   softmax over T per head (with mask -> -inf)
//   pass2: ctx[16 x D]   = P^T (16 x T) @ text[b] (T x D)    — WMMA fp32
// =====================================================================
__global__ void attn_kernel(const float* __restrict__ text,
                            const int* __restrict__ amask,
                            const float* __restrict__ qkp,
                            const float* __restrict__ sbias,
                            float* __restrict__ ctx) {
  const int b = blockIdx.x;
  const int lane = threadIdx.x & 31;
  const int wave = threadIdx.x >> 5;
  const int hl = lane & 15;
  const int koff = (lane >> 4) << 1;

  __shared__ float s_qkp[Dsz * 16];    // 48 KB  B-fragments for pass 1
  __shared__ float s_p[Tsz * 17];      // ~34.8 KB scores -> probabilities (pad 17 vs banks)
  __shared__ int   s_mask[Tsz];
  __shared__ float s_sb[Hsz];

  for (int i = threadIdx.x; i < Dsz * 16; i += 256)
    s_qkp[i] = qkp[(size_t)b * Dsz * 16 + i];
  for (int i = threadIdx.x; i < Tsz; i += 256)
    s_mask[i] = amask[(size_t)b * Tsz + i];
  if (threadIdx.x < Hsz) s_sb[threadIdx.x] = sbias[b * Hsz + threadIdx.x];
  __syncthreads();

  // ---- pass 1: scores ----
  const float* tb = text + (size_t)b * Tsz * Dsz;
  for (int i = 0; i < 4; ++i) {                 // 32 t-tiles / 8 waves
    const int t0 = (wave + 8 * i) * 16;
    const float* abase = tb + (size_t)(t0 + hl) * Dsz + koff;
    v8f acc = {0.f, 0.f, 0.f, 0.f, 0.f, 0.f, 0.f, 0.f};
    for (int k = 0; k < Dsz; k += 4) {
      v2f a = {abase[k], abase[k + 1]};
      const int kk = k + koff;
      v2f bb = {s_qkp[kk * 16 + hl], s_qkp[(kk + 1) * 16 + hl]};
      acc = wmma4(a, bb, acc);
    }
    const float sb = (hl < Hsz) ? s_sb[hl] : 0.f;
    const int rbase = t0 + ((lane >> 4) << 3);
    for (int j = 0; j < 8; ++j) {
      const int t = rbase + j;
      s_p[t * 17 + hl] = s_mask[t] ? (acc[j] + sb) : -__builtin_inff();
    }
  }
  __syncthreads();

  // ---- softmax per head: wave w handles head h=w ----
  {
    const int h = wave;  // 8 waves, 8 heads
    float m = -__builtin_inff();
    for (int t = lane; t < Tsz; t += 32) m = fmaxf(m, s_p[t * 17 + h]);
    for (int off = 16; off; off >>= 1) m = fmaxf(m, __shfl_xor(m, off, 32));
    float sum = 0.f;
    for (int t = lane; t < Tsz; t += 32) {
      float e = expf(s_p[t * 17 + h] - m);
      s_p[t * 17 + h] = e;
      sum += e;
    }
    for (int off = 16; off; off >>= 1) sum += __shfl_xor(sum, off, 32);
    const float inv = 1.0f / sum;
    for (int t = lane; t < Tsz; t += 32) s_p[t * 17 + h] *= inv;
  }
  __syncthreads();

  // ---- pass 2: ctx = P^T @ text[b] ----
  for (int i = 0; i < 6; ++i) {                 // 48 dd-tiles / 8 waves
    const int n0 = (wave * 6 + i) * 16;
    const float* bbase = tb + n0 + hl;
    v8f acc = {0.f, 0.f, 0.f, 0.f, 0.f, 0.f, 0.f, 0.f};
    for (int k = 0; k < Tsz; k += 4) {
      const int kk = k + koff;
      v2f a = {s_p[kk * 17 + hl], s_p[(kk + 1) * 17 + hl]};
      v2f bb = {bbase[(size_t)kk * Dsz], bbase[(size_t)(kk + 1) * Dsz]};
      acc = wmma4(a, bb, acc);
    }
    if (lane < 16) {  // rows M=0..7 are the real heads
      for (int j = 0; j < Hsz; ++j)
        ctx[((size_t)b * Hsz + j) * Dsz + n0 + lane] = acc[j];
    }
  }
}

// =====================================================================
// Generic fp32-WMMA GEMM: C[M x N] = A[M x K] @ W[N x K]^T + bias  (torch Linear)
// one 16x16 tile per wave; grid.x * 8 waves == (M/16)*(N/16)
// =====================================================================
__global__ void gemm_bias_kernel(const float* __restrict__ A,
                                 const float* __restrict__ W,
                                 const float* __restrict__ bias,
                                 float* __restrict__ C,
                                 int M, int N, int K) {
  const int lane = threadIdx.x & 31;
  const int wave = threadIdx.x >> 5;
  const int tile = blockIdx.x * (blockDim.x >> 5) + wave;
  const int ntn = N >> 4;
  const int m0 = (tile / ntn) << 4;
  const int n0 = (tile % ntn) << 4;
  if (m0 >= M) return;
  const int hl = lane & 15, koff = (lane >> 4) << 1;
  const float* abase = A + (size_t)(m0 + hl) * K + koff;
  const float* wbase = W + (size_t)(n0 + hl) * K + koff;
  v8f acc = {0.f, 0.f, 0.f, 0.f, 0.f, 0.f, 0.f, 0.f};
  for (int k = 0; k < K; k += 4) {
    v2f a = {abase[k], abase[k + 1]};
    v2f bb = {wbase[k], wbase[k + 1]};
    acc = wmma4(a, bb, acc);
  }
  const float bn = bias[n0 + hl];
  const int mrow = m0 + ((lane >> 4) << 3);
  for (int j = 0; j < 8; ++j)
    C[(size_t)(mrow + j) * N + n0 + hl] = acc[j] + bn;
}

// V projection: outA[b][f] = Wv[f,:] . ctx[b][f/128][:] + bv[f]   (M=256,N=1024,K=768)
__global__ void vproj_kernel(const float* __restrict__ ctx,
                             const float* __restrict__ Wv,
                             const float* __restrict__ bv,
                             float* __restrict__ outA) {
  const int lane = threadIdx.x & 31;
  const int wave = threadIdx.x >> 5;
  const int tile = blockIdx.x * (blockDim.x >> 5) + wave;
  const int ntn = Fsz >> 4;
  const int m0 = (tile / ntn) << 4;
  const int n0 = (tile % ntn) << 4;
  const int head = n0 >> 7;
  const int hl = lane & 15, koff = (lane >> 4) << 1;
  const float* abase = ctx + ((size_t)(m0 + hl) * Hsz + head) * Dsz + koff;
  const float* wbase = Wv + (size_t)(n0 + hl) * Dsz + koff;
  v8f acc = {0.f, 0.f, 0.f, 0.f, 0.f, 0.f, 0.f, 0.f};
  for (int k = 0; k < Dsz; k += 4) {
    v2f a = {abase[k], abase[k + 1]};
    v2f bb = {wbase[k], wbase[k + 1]};
    acc = wmma4(a, bb, acc);
  }
  const float bn = bv[n0 + hl];
  const int mrow = m0 + ((lane >> 4) << 3);
  for (int j = 0; j < 8; ++j)
    outA[(size_t)(mrow + j) * Fsz + n0 + hl] = acc[j] + bn;
}

// BatchNorm1d training stats over batch dim -> fused scale/shift per feature
__global__ void bn_stats_kernel(const float* __restrict__ h1,
                                const float* __restrict__ gamma,
                                const float* __restrict__ beta,
                                float* __restrict__ gs, float* __restrict__ gb) {
  const int j = blockIdx.x * blockDim.x + threadIdx.x;
  if (j >= HIDsz) return;
  float s = 0.f, s2 = 0.f;
  for (int b = 0; b < Bsz; ++b) {
    float v = h1[(size_t)b * HIDsz + j];
    s += v; s2 = fmaf(v, v, s2);
  }
  const float mu = s * (1.0f / Bsz);
  const float var = s2 * (1.0f / Bsz) - mu * mu;
  const float sc = gamma[j] / sqrtf(var + 1e-5f);
  gs[j] = sc;
  gb[j] = beta[j] - mu * sc;
}

// normalize + exact GELU
__global__ void bn_gelu_kernel(const float* __restrict__ h1,
                               const float* __restrict__ gs,
                               const float* __restrict__ gb,
                               float* __restrict__ g, int n) {
  const int i = blockIdx.x * blockDim.x + threadIdx.x;
  if (i >= n) return;
  const int j = i & (HIDsz - 1);
  const float y = fmaf(h1[i], gs[j], gb[j]);
  g[i] = 0.5f * y * (1.0f + erff(y * 0.70710678118654752f));
}

extern "C" void kernel_launch(void* const* d_in, const int* in_sizes, int n_in,
                              void* d_out, int out_size, void* d_ws, size_t ws_size,
                              hipStream_t stream) {
  (void)in_sizes; (void)n_in; (void)out_size; (void)ws_size;
  const float* x     = (const float*)d_in[0];
  const float* text  = (const float*)d_in[1];
  const int*   amask = (const int*)  d_in[2];
  const float* Wq = (const float*)d_in[3];  const float* bq = (const float*)d_in[4];
  const float* Wk = (const float*)d_in[5];  const float* bk = (const float*)d_in[6];
  const float* Wv = (const float*)d_in[7];  const float* bv = (const float*)d_in[8];
  const float* Wo = (const float*)d_in[9];  const float* bo = (const float*)d_in[10];
  const float* W1 = (const float*)d_in[11]; const float* b1 = (const float*)d_in[12];
  const float* gamma = (const float*)d_in[13]; const float* beta = (const float*)d_in[14];
  const float* W2 = (const float*)d_in[15]; const float* b2 = (const float*)d_in[16];
  float* out = (float*)d_out;

  // workspace layout (floats), ~22 MB total
  float* ws    = (float*)d_ws;
  float* qkp   = ws;                                   // 256*768*16
  float* sbias = qkp   + (size_t)Bsz * Dsz * 16;       // 256*8
  float* ctx   = sbias + (size_t)Bsz * Hsz;            // 256*8*768
  float* outA  = ctx   + (size_t)Bsz * Hsz * Dsz;      // 256*1024
  float* o2    = outA  + (size_t)Bsz * Fsz;            // 256*1024
  float* h1    = o2    + (size_t)Bsz * Fsz;            // 256*512
  float* gs    = h1    + (size_t)Bsz * HIDsz;          // 512
  float* gb    = gs    + HIDsz;                        // 512
  float* g     = gb    + HIDsz;                        // 256*512

  prep_qk_kernel<<<Bsz, 256, 0, stream>>>(x, Wq, bq, Wk, bk, qkp, sbias);
  attn_kernel<<<Bsz, 256, 0, stream>>>(text, amask, qkp, sbias, ctx);
  vproj_kernel<<<(16 * 64) / 8, 256, 0, stream>>>(ctx, Wv, bv, outA);
  gemm_bias_kernel<<<(16 * 64) / 8, 256, 0, stream>>>(outA, Wo, bo, o2, Bsz, Fsz, Fsz);
  gemm_bias_kernel<<<(16 * 32) / 8, 256, 0, stream>>>(o2, W1, b1, h1, Bsz, HIDsz, Fsz);
  bn_stats_kernel<<<2, 256, 0, stream>>>(h1, gamma, beta, gs, gb);
  bn_gelu_kernel<<<(Bsz * HIDsz + 255) / 256, 256, 0, stream>>>(h1, gs, gb, g, Bsz * HIDsz);
  gemm_bias_kernel<<<(16 * 16) / 8, 256, 0, stream>>>(g, W2, b2, out, Bsz, OUTsz, HIDsz);
}